// TransitionGCN_18330920419717
// MI455X (gfx1250) — compile-verified
//
#include <hip/hip_runtime.h>
#include <hip/hip_bf16.h>

#define B_DIM 4096
#define O_DIM 16
#define D_DIM 256
#define H_DIM 512

typedef __attribute__((ext_vector_type(2))) float v2f;
typedef __attribute__((ext_vector_type(8))) float v8f;

__device__ __forceinline__ v8f wmma_f32(v2f a, v2f b, v8f c) {
    // D = A(16x4) * B(4x16) + C, exact fp32 — v_wmma_f32_16x16x4_f32
    return __builtin_amdgcn_wmma_f32_16x16x4_f32(
        /*neg_a=*/false, a, /*neg_b=*/false, b,
        /*c_mod=*/(short)0, c, /*reuse_a=*/false, /*reuse_b=*/false);
}

// ---------------------------------------------------------------------------
// Kernel 1: xbar[b,d] = mean over o of states[b,o,d]   (64 MiB HBM read)
// ---------------------------------------------------------------------------
__global__ void __launch_bounds__(256)
k_reduce_states(const float* __restrict__ states, float* __restrict__ xbar) {
    const int b = blockIdx.x;
    const int d = threadIdx.x;  // 0..255
    const float* p = states + (size_t)b * (O_DIM * D_DIM) + d;
    float acc = 0.f;
#pragma unroll
    for (int o = 0; o < O_DIM; ++o) acc += p[o * D_DIM];
    xbar[(size_t)b * D_DIM + d] = acc * (1.0f / 16.0f);
}

// ---------------------------------------------------------------------------
// Kernel 2: h1 = relu(xbar @ W1[0:256,:] + W1[256+(action&3),:]/16 + b1)
// Each wave: 16x64 output tile (4 WMMA accumulators), K=256.
// ---------------------------------------------------------------------------
__global__ void __launch_bounds__(256)
k_gemm1(const float* __restrict__ xbar, const float* __restrict__ W1,
        const float* __restrict__ bias1, const int* __restrict__ action,
        float* __restrict__ h1) {
    const int K = D_DIM, N = H_DIM;
    const int wave = threadIdx.x >> 5;
    const int lane = threadIdx.x & 31;
    const int job  = blockIdx.x * 8 + wave;   // 2048 jobs
    const int m0 = (job >> 3) * 16;           // 256 m-tiles
    const int n0 = (job & 7) * 64;            // 8 n-groups
    const int laneM = lane & 15;
    const int laneK = (lane >> 4) * 2;        // 0 or 2

    v8f acc[4] = {};
    const float* arow = xbar + (size_t)(m0 + laneM) * K + laneK;
    for (int k = 0; k < K; k += 4) {
        v2f a = *(const v2f*)(arow + k);
        const float* bp = W1 + (size_t)(k + laneK) * N + laneM;
#pragma unroll
        for (int t = 0; t < 4; ++t) {
            v2f bf = { bp[n0 + t * 16], bp[N + n0 + t * 16] };
            acc[t] = wmma_f32(a, bf, acc[t]);
        }
    }
    const int mhalf = (lane >> 4) * 8;
#pragma unroll
    for (int r = 0; r < 8; ++r) {
        const int m = m0 + mhalf + r;
        const int c = action[m] & 3;
        const float* wtail = W1 + (size_t)(D_DIM + c) * N;
#pragma unroll
        for (int t = 0; t < 4; ++t) {
            const int n = n0 + t * 16 + laneM;
            float v = acc[t][r] + wtail[n] * 0.0625f + bias1[n];
            h1[(size_t)m * N + n] = fmaxf(v, 0.0f);
        }
    }
}

// ---------------------------------------------------------------------------
// Kernel 3: h2 = relu(layernorm(h1 @ W2 + b2) * g + b)
// One block = 16 rows x full 512 cols (8 waves x 64 cols); z staged in LDS,
// LayerNorm reduced in-block.
// ---------------------------------------------------------------------------
__global__ void __launch_bounds__(256)
k_gemm2_ln(const float* __restrict__ h1, const float* __restrict__ W2,
           const float* __restrict__ bias2, const float* __restrict__ ln_g,
           const float* __restrict__ ln_b, float* __restrict__ h2) {
    const int K = H_DIM, N = H_DIM;
    __shared__ float z[16 * H_DIM];        // 32 KB
    __shared__ float red_s[16][16];
    __shared__ float red_q[16][16];
    __shared__ float mu_s[16], rs_s[16];

    const int wave = threadIdx.x >> 5;
    const int lane = threadIdx.x & 31;
    const int m0 = blockIdx.x * 16;
    const int n0 = wave * 64;
    const int laneM = lane & 15;
    const int laneK = (lane >> 4) * 2;

    v8f acc[4] = {};
    const float* arow = h1 + (size_t)(m0 + laneM) * K + laneK;
    for (int k = 0; k < K; k += 4) {
        v2f a = *(const v2f*)(arow + k);
        const float* bp = W2 + (size_t)(k + laneK) * N + laneM;
#pragma unroll
        for (int t = 0; t < 4; ++t) {
            v2f bf = { bp[n0 + t * 16], bp[N + n0 + t * 16] };
            acc[t] = wmma_f32(a, bf, acc[t]);
        }
    }
    const int mhalf = (lane >> 4) * 8;
#pragma unroll
    for (int r = 0; r < 8; ++r) {
#pragma unroll
        for (int t = 0; t < 4; ++t) {
            const int n = n0 + t * 16 + laneM;
            z[(mhalf + r) * H_DIM + n] = acc[t][r] + bias2[n];
        }
    }
    __syncthreads();

    // LayerNorm over 512: 16 threads per row, 32 elems per thread
    const int row = threadIdx.x >> 4;
    const int seg = threadIdx.x & 15;
    {
        const float* zr = z + row * H_DIM + seg * 32;
        float s = 0.f, q = 0.f;
#pragma unroll
        for (int i = 0; i < 32; ++i) { float v = zr[i]; s += v; q += v * v; }
        red_s[row][seg] = s;
        red_q[row][seg] = q;
    }
    __syncthreads();
    if (threadIdx.x < 16) {
        float s = 0.f, q = 0.f;
#pragma unroll
        for (int i = 0; i < 16; ++i) { s += red_s[threadIdx.x][i]; q += red_q[threadIdx.x][i]; }
        const float mu = s * (1.0f / H_DIM);
        const float var = q * (1.0f / H_DIM) - mu * mu;
        mu_s[threadIdx.x] = mu;
        rs_s[threadIdx.x] = rsqrtf(var + 1e-5f);
    }
    __syncthreads();
    {
        const float mu = mu_s[row], rstd = rs_s[row];
        float* dst = h2 + (size_t)(m0 + row) * H_DIM;
        const float* zr = z + row * H_DIM;
#pragma unroll
        for (int i = 0; i < 32; ++i) {
            const int n = seg * 32 + i;
            float v = (zr[n] - mu) * rstd * ln_g[n] + ln_b[n];
            dst[n] = fmaxf(v, 0.0f);
        }
    }
}

// ---------------------------------------------------------------------------
// Kernel 4: out[b,o,:] = h2[b] @ W3 + b3 for all 16 o   (64 MiB HBM write)
// ---------------------------------------------------------------------------
__global__ void __launch_bounds__(256)
k_gemm3_bcast(const float* __restrict__ h2, const float* __restrict__ W3,
              const float* __restrict__ bias3, float* __restrict__ out) {
    const int K = H_DIM, N = D_DIM;
    const int wave = threadIdx.x >> 5;
    const int lane = threadIdx.x & 31;
    const int job  = blockIdx.x * 8 + wave;   // 1024 jobs
    const int m0 = (job >> 2) * 16;           // 256 m-tiles
    const int n0 = (job & 3) * 64;            // 4 n-groups
    const int laneM = lane & 15;
    const int laneK = (lane >> 4) * 2;

    v8f acc[4] = {};
    const float* arow = h2 + (size_t)(m0 + laneM) * K + laneK;
    for (int k = 0; k < K; k += 4) {
        v2f a = *(const v2f*)(arow + k);
        const float* bp = W3 + (size_t)(k + laneK) * N + laneM;
#pragma unroll
        for (int t = 0; t < 4; ++t) {
            v2f bf = { bp[n0 + t * 16], bp[N + n0 + t * 16] };
            acc[t] = wmma_f32(a, bf, acc[t]);
        }
    }
    const int mhalf = (lane >> 4) * 8;
#pragma unroll
    for (int r = 0; r < 8; ++r) {
        const int m = m0 + mhalf + r;
        float* obase = out + (size_t)m * (O_DIM * D_DIM);
#pragma unroll
        for (int t = 0; t < 4; ++t) {
            const int n = n0 + t * 16 + laneM;
            const float v = acc[t][r] + bias3[n];
#pragma unroll
            for (int o = 0; o < O_DIM; ++o)
                obase[o * D_DIM + n] = v;
        }
    }
}

extern "C" void kernel_launch(void* const* d_in, const int* in_sizes, int n_in,
                              void* d_out, int out_size, void* d_ws, size_t ws_size,
                              hipStream_t stream) {
    const float* states = (const float*)d_in[0];
    const int*   action = (const int*)d_in[1];
    const float* W1     = (const float*)d_in[2];
    const float* b1     = (const float*)d_in[3];
    const float* W2     = (const float*)d_in[4];
    const float* b2     = (const float*)d_in[5];
    const float* W3     = (const float*)d_in[6];
    const float* b3     = (const float*)d_in[7];
    const float* ln_g   = (const float*)d_in[8];
    const float* ln_b   = (const float*)d_in[9];
    float* out = (float*)d_out;

    // workspace: xbar 4MB | h1 8MB | h2 8MB  (20 MB total)
    float* xbar = (float*)d_ws;
    float* h1   = xbar + (size_t)B_DIM * D_DIM;
    float* h2   = h1   + (size_t)B_DIM * H_DIM;

    k_reduce_states<<<B_DIM, 256, 0, stream>>>(states, xbar);
    k_gemm1<<<(B_DIM / 16) * 8 / 8, 256, 0, stream>>>(xbar, W1, b1, action, h1);
    k_gemm2_ln<<<B_DIM / 16, 256, 0, stream>>>(h1, W2, b2, ln_g, ln_b, h2);
    k_gemm3_bcast<<<(B_DIM / 16) * 4 / 8, 256, 0, stream>>>(h2, W3, b3, out);
}